// FDTDSimulator_59992103190778
// MI455X (gfx1250) — compile-verified
//
#include <hip/hip_runtime.h>
#include <math.h>

typedef __attribute__((ext_vector_type(2))) float v2f;
typedef __attribute__((ext_vector_type(8))) float v8f;

#define NX     300
#define NYP    320      // pitched Y (floats); padding columns stay zero
#define NQP    80       // NYP / 4 float4 quads per row
#define STEPS  300
#define NPORTS 4

// port mask value folded with the 1/PORT_WIDTH normalization (for WMMA B)
__device__ __forceinline__ float pmaskv(int y, int p) {
    if (p >= NPORTS) return 0.0f;
    int c = 60 + 60 * p;                       // port centers 60,120,180,240
    return (y >= c - 10 && y < c + 10) ? (1.0f / 20.0f) : 0.0f;
}

__device__ __forceinline__ float dampS(int v) {  // graded absorber sigma(x)
    if (v < 10)  { float u = (float)(10 - v)  * 0.1f; return 0.5f * u * u * u; }
    if (v >= 290 && v < NX) { float u = (float)(v - 289) * 0.1f; return 0.5f * u * u * u; }
    return 0.0f;
}

__global__ __launch_bounds__(960)
void fdtd_persistent(const float* __restrict__ source,    // (2, 4)
                     const float* __restrict__ radius_m,  // (10, 10)
                     float* __restrict__ out,             // (2, 300, 4)
                     float* __restrict__ ws)
{
    const int b    = blockIdx.x;       // batch element
    const int q    = threadIdx.x;      // y-quad 0..79 (y = 4q..4q+3)
    const int ty   = threadIdx.y;      // 0..11 (rows x = ty + 12*i)
    const int flat = ty * NQP + q;     // 0..959
    const int y0   = 4 * q;

    float* base = ws + (size_t)b * 4u * NX * NYP;
    float*  Ez  = base;
    float*  Hx  = base + 1 * NX * NYP;
    float*  Hy  = base + 2 * NX * NYP;
    float*  iE  = base + 3 * NX * NYP;
    float4* Ez4 = (float4*)Ez;
    float4* Hx4 = (float4*)Hx;
    float4* Hy4 = (float4*)Hy;
    float4* iE4 = (float4*)iE;

    __shared__ float exTab[NX];                       // exp(-s(x)) (separable damp)
    __shared__ float r2s[100];                        // circle radius^2
    __shared__ __align__(16) float Edet[16 * NYP];    // 16 steps of Ez[270,:]

    const float COUR = 0.99f / sqrtf(2.0f);
    // OMEGA_T = 2*pi*DT/PERIOD = 2*pi*COURANT*DX/WAVELENGTH
    const float WT = (float)(2.0 * M_PI * (double)COUR * (25e-9 / 1550e-9));

    // ---- LDS tables ----
    if (flat < NX) exTab[flat] = expf(-dampS(flat));
    if (flat >= 512 && flat < 612) {
        float r = radius_m[flat - 512];
        r = (r < 0.3f) ? 0.0f : r;
        r *= 10.0f;
        r2s[flat - 512] = r * r;
    }
    for (int i = flat; i < 16 * NYP; i += 960) Edet[i] = 0.0f;

    // ---- per-thread y-constants (4 columns) ----
    float ey[4]; float sm[4]; int myPort = -1;
    #pragma unroll
    for (int c = 0; c < 4; ++c) {
        int yc = y0 + c;
        ey[c] = expf(-dampS(yc));
        sm[c] = 0.0f;
        if (yc >= 50 && yc < 250 && ((yc - 50) % 60) < 20) {
            myPort = (yc - 50) / 60;     // at most one port per quad
            sm[c] = 1.0f;
        }
    }
    float myPhase = 0.0f;
    if (myPort >= 0) myPhase = source[b * NPORTS + myPort] * 3.14159265358979323846f;

    __syncthreads();   // r2s ready

    // ---- init fields + permittivity (every call: determinism) ----
    for (int i = 0; i < 25; ++i) {
        int x    = ty + 12 * i;
        int idx4 = x * NQP + q;
        float4 z4; z4.x = z4.y = z4.z = z4.w = 0.0f;
        Ez4[idx4] = z4; Hx4[idx4] = z4; Hy4[idx4] = z4;
        float4 ie;
        float* iep = (float*)&ie;
        #pragma unroll
        for (int c = 0; c < 4; ++c) {
            int yc = y0 + c;
            float eps = 1.0f;
            if (yc < NX) {
                if ((x < 30 || x >= 270) && sm[c] > 0.0f) eps = 2.8f;  // waveguides
                if (x >= 50 && x < 250 && yc >= 50 && yc < 250) {      // circle field
                    float lx = (float)(x - 50), ly = (float)(yc - 50);
                    bool inside = false;
                    #pragma unroll 4
                    for (int cc = 0; cc < 100; ++cc) {
                        float cx = (float)((cc / 10) * 20 + 10);
                        float cy = (float)((cc % 10) * 20 + 10);
                        float dx = lx - cx, dy = ly - cy;
                        inside = inside || (dx * dx + dy * dy <= r2s[cc]);
                    }
                    eps = inside ? 1.0f : 2.8f;
                }
            }
            iep[c] = COUR / eps;
        }
        iE4[idx4] = ie;
    }
    __syncthreads();

    const bool active = (q < 75);            // quads covering y < 300
    const bool lastQ  = (q == 74);           // contains y = 299

    // ============================ time loop ============================
    for (int t = 0; t < STEPS; ++t) {
        // ---- H half-step: H -= damp * COUR * curl(E) (fwd diff, 0 at hi edge)
        if (active) {
            #pragma unroll 5
            for (int i = 0; i < 25; ++i) {
                int    x    = ty + 12 * i;
                int    idx4 = x * NQP + q;
                float  dx_  = exTab[x];
                float4 ez   = Ez4[idx4];
                float  nxt  = lastQ ? ez.w : Ez[x * NYP + y0 + 4];     // dEz/dy=0 @299
                float4 ezxp = (x < NX - 1) ? Ez4[idx4 + NQP] : ez;     // dEz/dx=0 @299
                float4 hx   = Hx4[idx4];
                float4 hy   = Hy4[idx4];
                hx.x = dx_ * ey[0] * (hx.x - COUR * (ez.y - ez.x));
                hx.y = dx_ * ey[1] * (hx.y - COUR * (ez.z - ez.y));
                hx.z = dx_ * ey[2] * (hx.z - COUR * (ez.w - ez.z));
                hx.w = dx_ * ey[3] * (hx.w - COUR * (nxt  - ez.w));
                hy.x = dx_ * ey[0] * (hy.x + COUR * (ezxp.x - ez.x));
                hy.y = dx_ * ey[1] * (hy.y + COUR * (ezxp.y - ez.y));
                hy.z = dx_ * ey[2] * (hy.z + COUR * (ezxp.z - ez.z));
                hy.w = dx_ * ey[3] * (hy.w + COUR * (ezxp.w - ez.w));
                Hx4[idx4] = hx;
                Hy4[idx4] = hy;
            }
        }
        __syncthreads();

        float amp = 0.0f;
        if (ty == 6 && myPort >= 0) amp = __sinf(WT * (float)t + myPhase);

        // ---- E half-step: E += damp * invEps * curl(H) (bwd diff, 0 at lo edge)
        if (active) {
            #pragma unroll 5
            for (int i = 0; i < 25; ++i) {
                int    x    = ty + 12 * i;
                int    idx4 = x * NQP + q;
                float  dx_  = exTab[x];
                float4 hy   = Hy4[idx4];
                float4 hym  = (x > 0) ? Hy4[idx4 - NQP] : hy;          // dHy/dx=0 @0
                float4 hx   = Hx4[idx4];
                float  hxp  = (q > 0) ? Hx[x * NYP + y0 - 1] : hx.x;   // dHx/dy=0 @0
                float4 ez   = Ez4[idx4];
                float4 ie   = iE4[idx4];
                ez.x = dx_ * ey[0] * (ez.x + ie.x * ((hy.x - hym.x) - (hx.x - hxp )));
                ez.y = dx_ * ey[1] * (ez.y + ie.y * ((hy.y - hym.y) - (hx.y - hx.x)));
                ez.z = dx_ * ey[2] * (ez.z + ie.z * ((hy.z - hym.z) - (hx.z - hx.y)));
                ez.w = dx_ * ey[3] * (ez.w + ie.w * ((hy.w - hym.w) - (hx.w - hx.z)));
                if (x == 30) {                                          // soft source
                    ez.x += amp * sm[0]; ez.y += amp * sm[1];
                    ez.z += amp * sm[2]; ez.w += amp * sm[3];
                }
                Ez4[idx4] = ez;
                if (x == 270)                                           // detector
                    ((float4*)Edet)[(t & 15) * NQP + q] = ez;
            }
        }
        __syncthreads();

        // ---- every 16 steps: wave 0 reduces intensities with V_WMMA_F32_16X16X4_F32
        // I[t', p] = sum_y Ez(t', y)^2 * mask(y, p)/20  ==  A(16xK) * B(Kx16), K=300
        if ((t & 15) == 15 || t == STEPS - 1) {
            if (flat < 32) {                     // full wave 0, EXEC all ones
                int  lane = flat;
                int  M    = lane & 15;           // step-in-block (A row)
                bool hi   = lane >= 16;
                int  Koff = hi ? 2 : 0;          // K slots per ISA 16x4 A layout
                int  N    = hi ? (lane - 16) : lane;   // B/D column
                v8f acc = {};
                for (int k = 0; k < 75; ++k) {   // K = 300 in chunks of 4
                    int   yb = 4 * k;
                    float e0 = Edet[M * NYP + yb + Koff];
                    float e1 = Edet[M * NYP + yb + Koff + 1];
                    v2f a;  a.x = e0 * e0;               a.y = e1 * e1;
                    v2f bm; bm.x = pmaskv(yb + Koff, N); bm.y = pmaskv(yb + Koff + 1, N);
                    acc = __builtin_amdgcn_wmma_f32_16x16x4_f32(
                              false, a, false, bm, (short)0, acc, false, false);
                }
                int t0 = t & ~15;
                if (N < NPORTS) {                // D: vgpr r -> M = r (+8 for hi half)
                    int Mb = hi ? 8 : 0;
                    #pragma unroll
                    for (int r = 0; r < 8; ++r) {
                        int tt = t0 + Mb + r;
                        if (tt < STEPS)
                            out[(b * STEPS + tt) * NPORTS + N] = acc[r];
                    }
                }
                // clear staging for the next (possibly partial) block
                for (int i2 = lane; i2 < 16 * NYP; i2 += 32) Edet[i2] = 0.0f;
            }
            __syncthreads();
        }
    }
}

extern "C" void kernel_launch(void* const* d_in, const int* in_sizes, int n_in,
                              void* d_out, int out_size, void* d_ws, size_t ws_size,
                              hipStream_t stream) {
    (void)in_sizes; (void)n_in; (void)out_size; (void)ws_size;
    const float* source   = (const float*)d_in[0];   // (2,4) fp32
    const float* radius_m = (const float*)d_in[1];   // (10,10) fp32
    float*       out      = (float*)d_out;           // (2,300,4) fp32
    float*       ws       = (float*)d_ws;            // 2 * 4 * 300 * 320 floats

    dim3 grid(2, 1, 1);        // one persistent workgroup per batch element
    dim3 block(NQP, 12, 1);    // 960 threads = 30 wave32s on one WGP
    fdtd_persistent<<<grid, block, 0, stream>>>(source, radius_m, out, ws);
}